// Net_11965778886927
// MI455X (gfx1250) — compile-verified
//
#include <hip/hip_runtime.h>
#include <hip/hip_bf16.h>
#include <math.h>

// MI455X / gfx1250, wave32. FP32 WMMA (V_WMMA_F32_16X16X4_F32) everywhere:
// ~2.6 GFLOP total, fully L2-resident -> latency bound, keep full precision
// for the cosine-sim argmax. k_dots stages W2 + y slabs into LDS via the
// Tensor Data Mover (tensor_load_to_lds, s_wait_tensorcnt) with double
// buffering, and reuses the gathered A fragments across all 16 column tiles.

typedef __attribute__((ext_vector_type(2))) float v2f;
typedef __attribute__((ext_vector_type(8))) float v8f;
typedef __attribute__((ext_vector_type(4))) unsigned int v4u;
typedef __attribute__((ext_vector_type(4))) int v4i;
typedef __attribute__((ext_vector_type(8))) int v8i;

#define BT   256   // B*T
#define DIM  64    // IDIM = ODIM = HDIM
#define MAXS 139

__device__ __constant__ int c_L[5] = {51, 57, 64, 70, 76};

static __device__ inline v8f wmma_f32(v2f a, v2f b, v8f c) {
  return __builtin_amdgcn_wmma_f32_16x16x4_f32(false, a, false, b, (short)0, c,
                                               false, false);
}

// LDS byte offset of a __shared__ object (flat addr truncates to LDS offset).
static __device__ inline unsigned lds_off(const void* p) {
  return (unsigned)(unsigned long long)p;
}

// Issue a TDM 2D tile load: global (rows x dim0 f32, row stride stride0) -> LDS.
static __device__ inline void tdm_load_2d(unsigned lds_addr, const void* gaddr,
                                          unsigned dim0, unsigned dim1,
                                          unsigned tile0, unsigned tile1,
                                          unsigned stride0) {
  unsigned long long ga = (unsigned long long)gaddr;
  v4u g0;
  g0[0] = 1u;                                             // count=1, user mode
  g0[1] = lds_addr;                                       // lds_addr
  g0[2] = (unsigned)(ga & 0xffffffffull);                 // global_addr lo
  g0[3] = (unsigned)((ga >> 32) & 0x1ffffffull) | (2u << 30); // hi | type=2
  v8i g1;
  g1[0] = (int)(2u << 16);                                // data_size = 4B
  g1[1] = (int)((dim0 & 0xffffu) << 16);                  // tensor_dim0 lo
  g1[2] = (int)((dim0 >> 16) | ((dim1 & 0xffffu) << 16)); // dim0 hi | dim1 lo
  g1[3] = (int)((dim1 >> 16) | (tile0 << 16));            // dim1 hi | tile_dim0
  g1[4] = (int)(tile1 & 0xffffu);                         // tile_dim1 (tile2=0)
  g1[5] = (int)stride0;                                   // tensor_dim0_stride
  g1[6] = 0;
  g1[7] = 0;
  v4i gz = (v4i)0;
#if defined(__clang_major__) && (__clang_major__ >= 23)
  __builtin_amdgcn_tensor_load_to_lds(g0, g1, gz, gz, (v8i)0, 0);
#else
  __builtin_amdgcn_tensor_load_to_lds(g0, g1, gz, gz, 0);
#endif
}

// ---------------------------------------------------------------------------
// Generic NT GEMM: C[m,n] = sum_k A[m,k]*B[n,k] (+ bias[n]*bias_scale) (*mask)
// One wave per 16x16 tile; K-loop in steps of 4 (f32 WMMA).
// ---------------------------------------------------------------------------
__global__ void k_gemm_nt(const float* __restrict__ A, const float* __restrict__ Bm,
                          float* __restrict__ C, int M, int N, int K,
                          const float* __restrict__ bias, float bias_scale,
                          const int* __restrict__ mask) {
  int ntn  = N >> 4;
  int tm   = blockIdx.x / ntn;
  int tn   = blockIdx.x % ntn;
  int lane = threadIdx.x & 31;
  int lrow = lane & 15;
  int koff = (lane >> 4) << 1;

  const float* arow = A  + (size_t)(tm * 16 + lrow) * K;
  const float* brow = Bm + (size_t)(tn * 16 + lrow) * K;

  v8f acc = {};
  for (int kb = 0; kb < K; kb += 4) {
    v2f a, b;
    a.x = arow[kb + koff];  a.y = arow[kb + koff + 1];
    b.x = brow[kb + koff];  b.y = brow[kb + koff + 1];
    acc = wmma_f32(a, b, acc);
  }

  int n  = tn * 16 + lrow;
  float bv = bias ? bias[n] * bias_scale : 0.f;
#pragma unroll
  for (int v = 0; v < 8; ++v) {
    int m = tm * 16 + v + ((lane >> 4) << 3);
    float val = acc[v] + bv;
    if (mask) val *= (float)mask[m * N + n];
    C[(size_t)m * N + n] = val;
  }
}

// ---------------------------------------------------------------------------
// dots[(h*S+s), bt] = sum_i G_theta[h,s,i] * y[bt,i]
// G[h,s,i] = W2[src[s+i-63], h] if 0 <= s+i-63 < L else 0  (interp + pad)
// One wave owns a 16-row stripe x all 256 columns:
//   - TDM stages W2 (16KB) and y in 64-row slabs (2x16KB, double buffered)
//   - A fragments gathered once from LDS into 32 VGPRs, reused for 16 tiles
//   - 256 WMMAs per wave.
// ---------------------------------------------------------------------------
__global__ void __launch_bounds__(32) k_dots(const float* __restrict__ W2,
                                             const float* __restrict__ y,
                                             float* __restrict__ dots,
                                             int L, int S) {
  __shared__ float s_w2[DIM * DIM];      // 16 KB
  __shared__ float s_y[2][64 * DIM];     // 2 x 16 KB, double buffered
  __shared__ int   s_src[80];

  int lane = threadIdx.x & 31;
  int lrow = lane & 15;
  int koff = (lane >> 4) << 1;

  // Stage W2 and the first y slab via the Tensor Data Mover.
  tdm_load_2d(lds_off(s_w2), W2, DIM, DIM, DIM, DIM, DIM);
  tdm_load_2d(lds_off(s_y[0]), y, DIM, 64, DIM, 64, DIM);

  for (int l = lane; l < L; l += 32) s_src[l] = (l * DIM) / L;

  int r = blockIdx.x * 16 + lrow;        // this lane's A row in (h,s) space
  int h = r / S;
  int s = r % S;

  // In-order TDM completion: <=1 outstanding means W2 is resident.
  __builtin_amdgcn_s_wait_tensorcnt(1);
  asm volatile("" ::: "memory");

  float ax[16], ay[16];
#pragma unroll
  for (int kb = 0; kb < 16; ++kb) {
    int p0 = s + kb * 4 + koff - 63;
    int p1 = p0 + 1;
    ax[kb] = (p0 >= 0 && p0 < L) ? s_w2[s_src[p0] * DIM + h] : 0.f;
    ay[kb] = (p1 >= 0 && p1 < L) ? s_w2[s_src[p1] * DIM + h] : 0.f;
  }

#pragma unroll
  for (int cg = 0; cg < 4; ++cg) {       // 4 slabs of 64 bt columns
    if (cg < 3)
      tdm_load_2d(lds_off(s_y[(cg + 1) & 1]),
                  y + (size_t)(cg + 1) * 64 * DIM, DIM, 64, DIM, 64, DIM);
    if (cg < 3)
      __builtin_amdgcn_s_wait_tensorcnt(1);  // slab cg done (in-order)
    else
      __builtin_amdgcn_s_wait_tensorcnt(0);
    asm volatile("" ::: "memory");

    const float* yb = s_y[cg & 1];
    for (int tn = 0; tn < 4; ++tn) {
      v8f acc = {};
#pragma unroll
      for (int kb = 0; kb < 16; ++kb) {
        int k0 = kb * 4 + koff;
        v2f a, b;
        a.x = ax[kb];
        a.y = ay[kb];
        b.x = yb[(tn * 16 + lrow) * DIM + k0];
        b.y = yb[(tn * 16 + lrow) * DIM + k0 + 1];
        acc = wmma_f32(a, b, acc);
      }
      int bt = cg * 64 + tn * 16 + lrow;
#pragma unroll
      for (int v = 0; v < 8; ++v) {
        int row = blockIdx.x * 16 + v + ((lane >> 4) << 3);
        dots[(size_t)row * BT + bt] = acc[v];
      }
    }
  }
}

// ---------------------------------------------------------------------------
// Per-theta (b,t)-independent window statistics:
//   nrm2G[h*S+s]  = sum_i G[h,s,i]^2
//   crossGB[h*S+s]= sum_i G[h,s,i]*b2p[s+i]
//   nrm2B[s]      = sum_i b2p[s+i]^2
// ---------------------------------------------------------------------------
__global__ void k_stats(const float* __restrict__ W2, const float* __restrict__ b2,
                        float* __restrict__ nrm2G, float* __restrict__ crossGB,
                        float* __restrict__ nrm2B, int L, int S) {
  int tid = blockIdx.x * blockDim.x + threadIdx.x;
  if (tid < DIM * S) {
    int h = tid / S, s = tid % S;
    float n2 = 0.f, cb = 0.f;
    for (int i = 0; i < DIM; ++i) {
      int q = s + i - 63;
      if (q >= 0 && q < L) {
        int src = (q * DIM) / L;
        float g  = W2[src * DIM + h];
        float bb = b2[src];
        n2 += g * g;
        cb += g * bb;
      }
    }
    nrm2G[tid] = n2;
    crossGB[tid] = cb;
  }
  if (tid < S) {
    float nb = 0.f;
    for (int i = 0; i < DIM; ++i) {
      int q = tid + i - 63;
      if (q >= 0 && q < L) { float bb = b2[(q * DIM) / L]; nb += bb * bb; }
    }
    nrm2B[tid] = nb;
  }
}

// D[s*BT+bt] = sum_i b2p[s+i] * y[bt,i]
__global__ void k_bdot(const float* __restrict__ b2, const float* __restrict__ y,
                       float* __restrict__ Dv, int L, int S) {
  int tid = blockIdx.x * blockDim.x + threadIdx.x;
  if (tid >= S * BT) return;
  int s = tid / BT, bt = tid % BT;
  const float* yr = y + (size_t)bt * DIM;
  float acc = 0.f;
  for (int i = 0; i < DIM; ++i) {
    int q = s + i - 63;
    if (q >= 0 && q < L) acc += b2[(q * DIM) / L] * yr[i];
  }
  Dv[tid] = acc;
}

__global__ void k_qnorm(const float* __restrict__ y, float* __restrict__ qn) {
  int bt = threadIdx.x;
  if (bt >= BT) return;
  float s = 0.f;
  for (int i = 0; i < DIM; ++i) { float v = y[bt * DIM + i]; s += v * v; }
  qn[bt] = sqrtf(s);
}

__global__ void k_init(float* __restrict__ best_sim) {
  int tid = blockIdx.x * blockDim.x + threadIdx.x;
  if (tid < BT * DIM) best_sim[tid] = -3.402823e38f;
}

// ---------------------------------------------------------------------------
// Per (bt,h): scan all shifts of current theta, reconstruct cosine sim from the
// rank-1 decomposition, track running best (strict > == JAX first-argmax +
// cross-theta strict-< update semantics).
// ---------------------------------------------------------------------------
__global__ void k_update(const float* __restrict__ hbuf, const float* __restrict__ qn,
                         const float* __restrict__ dots, const float* __restrict__ Dv,
                         const float* __restrict__ nrm2G, const float* __restrict__ crossGB,
                         const float* __restrict__ nrm2B,
                         float* __restrict__ best_sim, int* __restrict__ best_ts,
                         int theta_idx, int S) {
  int tid = blockIdx.x * blockDim.x + threadIdx.x;   // tid = h*BT + bt
  if (tid >= BT * DIM) return;
  int bt = tid % BT;
  int h  = tid / BT;
  float hv = hbuf[bt * DIM + h];
  float q  = qn[bt];
  float bsim = -3.402823e38f;
  int bs = 0;
  for (int s = 0; s < S; ++s) {
    float dotv = hv * dots[(size_t)(h * S + s) * BT + bt] + Dv[s * BT + bt];
    float wn2  = hv * hv * nrm2G[h * S + s] + 2.f * hv * crossGB[h * S + s] + nrm2B[s];
    float den  = q * sqrtf(wn2);
    float sim  = (den > 0.f) ? (dotv / den) : 0.f;
    if (sim > bsim) { bsim = sim; bs = s; }
  }
  if (bsim > best_sim[tid]) {
    best_sim[tid] = bsim;
    best_ts[tid]  = (theta_idx << 10) | bs;
  }
}

// winsum[bt,i] = sum_h ( hv*G_theta*[h, s*+i-63] + b2p[s*+i-63] )
__global__ void k_winsum(const float* __restrict__ hbuf, const int* __restrict__ best_ts,
                         const float* __restrict__ W2, const float* __restrict__ b2,
                         float* __restrict__ winsum) {
  int bt = blockIdx.x;
  int i  = threadIdx.x;            // 64 threads
  float acc = 0.f;
  for (int h = 0; h < DIM; ++h) {
    int ts = best_ts[h * BT + bt];
    int th = ts >> 10;
    int s  = ts & 1023;
    int L  = c_L[th];
    float hv = hbuf[bt * DIM + h];
    int q = s + i - 63;
    if (q >= 0 && q < L) {
      int src = (q * DIM) / L;
      acc += hv * W2[src * DIM + h] + b2[src];
    }
  }
  winsum[bt * DIM + i] = acc;
}

// masked MSE: loss = sum_{y!=0} (xf-y)^2 / count(y!=0)
__global__ void k_loss(const float* __restrict__ xf, const float* __restrict__ y,
                       float* __restrict__ loss) {
  __shared__ float ssum[256];
  __shared__ float scnt[256];
  int t = threadIdx.x;
  float s = 0.f, c = 0.f;
  for (int idx = t; idx < BT * DIM; idx += 256) {
    float yv = y[idx];
    if (yv != 0.f) { float d = xf[idx] - yv; s += d * d; c += 1.f; }
  }
  ssum[t] = s; scnt[t] = c;
  __syncthreads();
  for (int off = 128; off > 0; off >>= 1) {
    if (t < off) { ssum[t] += ssum[t + off]; scnt[t] += scnt[t + off]; }
    __syncthreads();
  }
  if (t == 0) loss[0] = ssum[0] / scnt[0];
}

// ---------------------------------------------------------------------------
extern "C" void kernel_launch(void* const* d_in, const int* in_sizes, int n_in,
                              void* d_out, int out_size, void* d_ws, size_t ws_size,
                              hipStream_t stream) {
  const float* x    = (const float*)d_in[0];
  const float* y    = (const float*)d_in[1];
  const int*   mask = (const int*)d_in[2];
  const float* W1   = (const float*)d_in[3];
  const float* b1   = (const float*)d_in[4];
  const float* W2   = (const float*)d_in[5];
  const float* b2   = (const float*)d_in[6];
  float* out = (float*)d_out;                 // [0]=loss, [1..16384]=xf

  float* ws       = (float*)d_ws;
  float* hbuf     = ws;                          // 16384
  float* qn       = hbuf + BT * DIM;             // 256
  float* best_sim = qn + BT;                     // 16384
  int*   best_ts  = (int*)(best_sim + BT * DIM); // 16384
  float* winsum   = (float*)(best_ts + BT * DIM);// 16384
  float* gbuf     = winsum + BT * DIM;           // 16384
  float* nrm2G    = gbuf + BT * DIM;             // 64*MAXS
  float* crossGB  = nrm2G + DIM * MAXS;          // 64*MAXS
  float* nrm2B    = crossGB + DIM * MAXS;        // MAXS (pad to 160)
  float* Dv       = nrm2B + 160;                 // MAXS*BT
  float* dots     = Dv + MAXS * BT;              // 64*MAXS*BT (~9.1 MB)

  static const int Ls[5] = {51, 57, 64, 70, 76};

  // h = (x @ W1^T + b1) * mask    (WMMA)
  k_gemm_nt<<<(BT / 16) * (DIM / 16), 32, 0, stream>>>(x, W1, hbuf, BT, DIM, DIM,
                                                       b1, 1.f, mask);
  k_qnorm<<<1, 256, 0, stream>>>(y, qn);
  k_init<<<64, 256, 0, stream>>>(best_sim);

  for (int th = 0; th < 5; ++th) {
    int L = Ls[th], S = L + 63;
    int hs = DIM * S;
    k_stats<<<(hs + 255) / 256, 256, 0, stream>>>(W2, b2, nrm2G, crossGB, nrm2B, L, S);
    k_bdot<<<(S * BT + 255) / 256, 256, 0, stream>>>(b2, y, Dv, L, S);
    k_dots<<<4 * S, 32, 0, stream>>>(W2, y, dots, L, S);  // WMMA + TDM
    k_update<<<64, 256, 0, stream>>>(hbuf, qn, dots, Dv, nrm2G, crossGB, nrm2B,
                                     best_sim, best_ts, th, S);
  }

  k_winsum<<<BT, DIM, 0, stream>>>(hbuf, best_ts, W2, b2, winsum);
  // g  = winsum @ W1^T + 64*b1 ; out = g @ W2^T + 64*b2   (WMMA)
  k_gemm_nt<<<(BT / 16) * (DIM / 16), 32, 0, stream>>>(winsum, W1, gbuf, BT, DIM, DIM,
                                                       b1, 64.f, nullptr);
  k_gemm_nt<<<(BT / 16) * (DIM / 16), 32, 0, stream>>>(gbuf, W2, out + 1, BT, DIM, DIM,
                                                       b2, 64.f, nullptr);
  k_loss<<<1, 256, 0, stream>>>(out + 1, y, out);
}